// MyModel_61933428410243
// MI455X (gfx1250) — compile-verified
//
#include <hip/hip_runtime.h>

// ---------------------------------------------------------------------------
// LSTM scan (T=2048, B=2048, IN=5, H=10) for gfx1250 (MI455X, wave32, WMMA).
//
// One wave owns a 16-row batch tile for the whole time loop (recurrence is
// independent per batch row -> no cross-wave sync at all).
// Per step: gates[16x64] = [x_t | pad | h | pad][16x20] @ Wcat[20x64] + bias
// done as 5 K-steps x 4 N-tiles of V_WMMA_F32_16X16X4_F32 (exact fp32 math).
// Gate groups i/f/g/o are padded to separate 16-col N-tiles so the LSTM cell
// update is register-aligned elementwise VALU (lane = hidden idx, reg = row).
// h feedback transposed D-layout -> A-layout through wave-private LDS
// (in-order DS, no barriers). Weights live in registers in B-layout.
// Bias lives in persistent C-layout registers consumed by the first WMMA.
// Output stores are non-temporal (168MB write-once; keep L2 for the x stream).
// ---------------------------------------------------------------------------

typedef __attribute__((ext_vector_type(2))) float v2f;
typedef __attribute__((ext_vector_type(8))) float v8f;

#define T_STEPS 2048
#define BATCH   2048
#define IN_DIM  5
#define HID     10

// W_cat[k][n]: k in [0,20): rows 0-4 = W_ih columns, 5-7 = zero pad,
// 8-17 = W_hh columns, 18-19 = zero pad. n = nt*16 + col, gate row = nt*10+col
// (col >= 10 is zero padding). Gate order nt = 0:i 1:f 2:g 3:o (torch order).
__device__ __forceinline__ float wcat(const float* __restrict__ Wih,
                                      const float* __restrict__ Whh,
                                      int k, int col, int nt) {
  if (col >= 10) return 0.0f;
  const int row = nt * 10 + col;           // row of W_ih / W_hh ([40 x K])
  if (k < 5)              return Wih[row * 5 + k];
  if (k >= 8 && k < 18)   return Whh[row * 10 + (k - 8)];
  return 0.0f;
}

__device__ __forceinline__ float fast_sigmoid(float x) {
  // 1 / (1 + e^-x); x->+inf: exp->0 -> 1; x->-inf: exp->inf -> rcp -> 0. Safe.
  return __builtin_amdgcn_rcpf(1.0f + __expf(-x));
}
__device__ __forceinline__ float fast_tanh(float x) {
  // tanh(x) = sign(x) * (1 - t) / (1 + t), t = e^{-2|x|} in (0,1]; no overflow.
  const float t = __expf(-2.0f * __builtin_fabsf(x));
  const float r = (1.0f - t) * __builtin_amdgcn_rcpf(1.0f + t);
  return __builtin_copysignf(r, x);
}

__global__ __launch_bounds__(256) void lstm_scan_wmma(
    const float* __restrict__ x,    // [T, B, IN]
    const float* __restrict__ hx0,  // [B, H]
    const float* __restrict__ cx0,  // [B, H]
    const float* __restrict__ Wih,  // [4H, IN]
    const float* __restrict__ Whh,  // [4H, H]
    const float* __restrict__ bih,  // [4H]
    const float* __restrict__ bhh,  // [4H]
    float* __restrict__ out)        // [T*B, H]
{
  __shared__ float hbuf[8][16][17];        // per-wave h tile, padded stride

  const int lane = threadIdx.x & 31;
  const int wib  = threadIdx.x >> 5;       // wave in block (0..7)
  const int wave = blockIdx.x * 8 + wib;   // 0..127
  const int b0   = wave * 16;              // batch tile base
  const int hi   = lane >> 4;              // half-wave (A/B layout K select)
  const int lm   = lane & 15;              // M row (A) / N col (B,C,D)
  const float fhi = hi ? 0.0f : 1.0f;      // mask for padded K lanes

  // ---- weights into B-matrix layout registers: Bw[kstep][ntile] -----------
  // B layout (4x16 f32): v.x = row K = 4*js + hi*2, v.y = K+1; col = lm.
  v2f Bw[5][4];
#pragma unroll
  for (int js = 0; js < 5; ++js) {
#pragma unroll
    for (int nt = 0; nt < 4; ++nt) {
      const int k0 = js * 4 + hi * 2;
      Bw[js][nt].x = wcat(Wih, Whh, k0,     lm, nt);
      Bw[js][nt].y = wcat(Wih, Whh, k0 + 1, lm, nt);
    }
  }

  // bias in persistent C/D-layout registers (row-independent splat per tile);
  // consumed as the C operand of the first WMMA each step (dest != C, legal).
  v8f cbias[4];
#pragma unroll
  for (int nt = 0; nt < 4; ++nt) {
    const float bv = (lm < 10) ? (bih[nt * 10 + lm] + bhh[nt * 10 + lm]) : 0.0f;
    v8f a = {bv, bv, bv, bv, bv, bv, bv, bv};
    cbias[nt] = a;
  }

  // ---- state init: c in D layout regs, h0 into LDS [row][hidden] ----------
  // D layout: reg v -> row v (lanes 0-15) / row v+8 (lanes 16-31); col = lm.
  float c[8];
#pragma unroll
  for (int v = 0; v < 8; ++v) {
    const int b = b0 + v + hi * 8;
    c[v] = (lm < 10) ? cx0[b * HID + lm] : 0.0f;
    hbuf[wib][v + hi * 8][lm] = (lm < 10) ? hx0[b * HID + lm] : 0.0f;
  }

  // ---- x prefetch for t = 0 (A layout: .x = K even, .y = K odd) -----------
  const int bx = b0 + lm;                  // batch row for A loads
  float xa0 = x[(0 * BATCH + bx) * IN_DIM + hi * 2 + 0];        // K = 0 / 2
  float xa1 = x[(0 * BATCH + bx) * IN_DIM + hi * 2 + 1];        // K = 1 / 3
  float xa2 = fhi * x[(0 * BATCH + bx) * IN_DIM + 4];           // K = 4 / pad

  for (int t = 0; t < T_STEPS; ++t) {
    v8f acc[4];

    // K-step 0: x part, C = persistent bias registers
    v2f A;
    A.x = xa0; A.y = xa1;
#pragma unroll
    for (int nt = 0; nt < 4; ++nt)
      acc[nt] = __builtin_amdgcn_wmma_f32_16x16x4_f32(
          false, A, false, Bw[0][nt], (short)0, cbias[nt], false, false);
    // K-step 1 (K=4 valid, K=5..7 zero pad)
    A.x = xa2; A.y = 0.0f;
#pragma unroll
    for (int nt = 0; nt < 4; ++nt)
      acc[nt] = __builtin_amdgcn_wmma_f32_16x16x4_f32(
          false, A, false, Bw[1][nt], (short)0, acc[nt], false, false);

    // software-pipeline: issue next step's x loads + distant prefetch now.
    // All loads unconditional (hi lanes re-read a valid address, then mask).
    const int tn = (t + 1 < T_STEPS) ? (t + 1) : t;
    const float nxa0 = x[(tn * BATCH + bx) * IN_DIM + hi * 2 + 0];
    const float nxa1 = x[(tn * BATCH + bx) * IN_DIM + hi * 2 + 1];
    const float nxa2 = fhi * x[(tn * BATCH + bx) * IN_DIM + 4];
    const int tp = (t + 8 < T_STEPS) ? (t + 8) : (T_STEPS - 1);
    __builtin_prefetch(&x[((size_t)tp * BATCH + bx) * IN_DIM], 0, 1);

    // K-steps 2-4: h part, A fetched from wave-private LDS (transposed feed)
#pragma unroll
    for (int js = 0; js < 3; ++js) {
      v2f Ah;
      Ah.x = hbuf[wib][lm][js * 4 + hi * 2 + 0];
      Ah.y = hbuf[wib][lm][js * 4 + hi * 2 + 1];
#pragma unroll
      for (int nt = 0; nt < 4; ++nt)
        acc[nt] = __builtin_amdgcn_wmma_f32_16x16x4_f32(
            false, Ah, false, Bw[js + 2][nt], (short)0, acc[nt], false, false);
    }

    // elementwise LSTM cell update at full EXEC (no per-v predication)
    float hn[8];
#pragma unroll
    for (int v = 0; v < 8; ++v) {
      const float ig = fast_sigmoid(acc[0][v]);
      const float fg = fast_sigmoid(acc[1][v]);
      const float gg = fast_tanh(acc[2][v]);
      const float og = fast_sigmoid(acc[3][v]);
      const float cn = fg * c[v] + ig * gg;
      c[v] = cn;
      hn[v] = og * fast_tanh(cn);
      hbuf[wib][v + hi * 8][lm] = hn[v];   // feed next step (in-order DS)
    }

    // single predicated region for all 8 output stores; non-temporal (write-
    // once stream, keep L2 for x)
    if (lm < 10) {
      float* orow = out + ((size_t)t * BATCH + (size_t)(b0 + hi * 8)) * HID + lm;
#pragma unroll
      for (int v = 0; v < 8; ++v)
        __builtin_nontemporal_store(hn[v], orow + (size_t)v * HID);
    }

    xa0 = nxa0; xa1 = nxa1; xa2 = nxa2;
  }
}

extern "C" void kernel_launch(void* const* d_in, const int* in_sizes, int n_in,
                              void* d_out, int out_size, void* d_ws, size_t ws_size,
                              hipStream_t stream) {
  const float* x   = (const float*)d_in[0];
  const float* hx0 = (const float*)d_in[1];
  const float* cx0 = (const float*)d_in[2];
  const float* Wih = (const float*)d_in[3];
  const float* Whh = (const float*)d_in[4];
  const float* bih = (const float*)d_in[5];
  const float* bhh = (const float*)d_in[6];
  float* out = (float*)d_out;

  dim3 grid(BATCH / (16 * 8));   // 16 blocks of 8 waves -> 128 batch tiles
  dim3 block(256);
  hipLaunchKernelGGL(lstm_scan_wmma, grid, block, 0, stream,
                     x, hx0, cx0, Wih, Whh, bih, bhh, out);
}